// CenterLoss_42322607735528
// MI455X (gfx1250) — compile-verified
//
#include <hip/hip_runtime.h>
#include <hip/hip_bf16.h>
#include <stdint.h>

#define NCLS 1000
#define DIM  256
#define MOM  0.99f
#define EPSV 1e-6f
#define K2_WAVES 2
#define PITCH 260   // 256 + 4 pad -> conflict-free ds_load_b64 in WMMA phase

typedef float f4  __attribute__((ext_vector_type(4)));
typedef float f2  __attribute__((ext_vector_type(2)));
typedef float v8f __attribute__((ext_vector_type(8)));

// ---------------- ws layout (element units) ----------------
// [0, N)            : packed u32  = (pmark<<16) | target
// [N, N+C*D)        : centers_new (16B-aligned f32; d_out copy is misaligned by 1)
// [.., +C)          : clean counts per class (float)
// [.., +C)          : per-class sum of clean distances (atomic f32)
// [.., +2)          : {poison sq-err sum, poison count} (atomic f32)

__global__ void k_init(const int* __restrict__ targets, const int* __restrict__ pmarks,
                       uint32_t* __restrict__ packed, float* __restrict__ dists,
                       float* __restrict__ lossacc, int N) {
  int i = blockIdx.x * blockDim.x + threadIdx.x;
  if (i < N) packed[i] = (uint32_t)targets[i] | ((uint32_t)(pmarks[i] != 0) << 16);
  if (i < NCLS) dists[i] = 0.0f;
  if (i < 2) lossacc[i] = 0.0f;
}

// One wave per class: ballot-scan the packed array, register-accumulate clean
// rows (lane owns dims [8*lane, 8*lane+8)), EMA-update the center. No atomics.
__global__ void __launch_bounds__(32) k_center(
    const uint32_t* __restrict__ packed, const float* __restrict__ features,
    const float* __restrict__ center, float* __restrict__ centers_new,
    float* __restrict__ counts, int N) {
  const uint32_t c = blockIdx.x;
  const int lane = threadIdx.x;
  f4 a0 = (f4)0.0f, a1 = (f4)0.0f;
  int cnt = 0;
  for (int s0 = 0; s0 < N; s0 += 32) {
    uint32_t pw = packed[s0 + lane];            // clean match <=> pw == c
    uint32_t mask = __builtin_amdgcn_ballot_w32(pw == c);
    if (mask) {
      cnt += __builtin_popcount(mask);
      do {
        int b = __builtin_ctz(mask);
        mask &= mask - 1;
        long long base = (long long)(s0 + b) * DIM + lane * 8;
        a0 += *(const f4*)(features + base);
        a1 += *(const f4*)(features + base + 4);
      } while (mask);
    }
  }
  float inv = 1.0f / (float)(cnt > 0 ? cnt : 1);
  long long cb = (long long)c * DIM + lane * 8;
  f4 c0 = *(const f4*)(center + cb);
  f4 c1 = *(const f4*)(center + cb + 4);
  f4 o0 = c0, o1 = c1;
  if (cnt > 0) {
    float w = inv * (1.0f - MOM);
    o0 = c0 * MOM + a0 * w;
    o1 = c1 * MOM + a1 * w;
  }
  *(f4*)(centers_new + cb)     = o0;
  *(f4*)(centers_new + cb + 4) = o1;
  if (lane == 0) counts[c] = (float)cnt;
}

// Per wave: stage a 16-row diff tile in LDS, then 64x V_WMMA_F32_16X16X4_F32
// with A==B (symmetric Gram) so diag(acc) = sum_k diff[m,k]^2 per sample.
// Clean rows get +EPS (matches torch.pairwise_distance eps-on-diff); poison
// rows get no EPS so the same diagonal is the MSE numerator.
__global__ void __launch_bounds__(32 * K2_WAVES) k_dist(
    const uint32_t* __restrict__ packed, const float* __restrict__ features,
    const float* __restrict__ centers_new, float* __restrict__ dists,
    float* __restrict__ lossacc, int N) {
  __shared__ float tilebuf[K2_WAVES][16 * PITCH];
  const int lane = threadIdx.x & 31;
  const int wv   = threadIdx.x >> 5;
  const int s0   = (blockIdx.x * K2_WAVES + wv) * 16;
  if (s0 >= N) return;
  float* my = tilebuf[wv];

  // ---- stage diff tile (coalesced f4 global loads, f4 LDS stores) ----
  if (s0 + 16 < N)  // warm next tile's feature lines
    __builtin_prefetch(features + (long long)(s0 + 16) * DIM + lane * 8, 0, 0);
  for (int r = 0; r < 16; ++r) {
    int s = s0 + r;
    f4 d0 = (f4)0.0f, d1 = (f4)0.0f;
    if (s < N) {
      uint32_t pw = packed[s];
      int   t = (int)(pw & 0xffffu);
      float e = (pw >> 16) ? 0.0f : EPSV;
      long long fb = (long long)s * DIM + lane * 8;
      long long cb = (long long)t * DIM + lane * 8;
      f4 f0 = *(const f4*)(features + fb);
      f4 f1 = *(const f4*)(features + fb + 4);
      f4 c0 = *(const f4*)(centers_new + cb);
      f4 c1 = *(const f4*)(centers_new + cb + 4);
      d0 = f0 - c0 + e;
      d1 = f1 - c1 + e;
    }
    *(f4*)(my + r * PITCH + lane * 8)     = d0;
    *(f4*)(my + r * PITCH + lane * 8 + 4) = d1;
  }

  // ---- Gram via WMMA: A layout == B layout for the symmetric case.
  // Lane L supplies row (L&15), K columns k0 + 2*(L>>4) + {0,1}.
  const int row  = lane & 15;
  const int coff = (lane >> 4) * 2;
  v8f acc = {};
  for (int k0 = 0; k0 < DIM; k0 += 4) {
    f2 ab = *(const f2*)(my + row * PITCH + k0 + coff);   // ds_load_b64, bank-conflict-free
    acc = __builtin_amdgcn_wmma_f32_16x16x4_f32(
        /*neg_a=*/false, ab, /*neg_b=*/false, ab,
        /*c_mod=*/(short)0, acc, /*reuse_a=*/false, /*reuse_b=*/false);
  }

  // ---- diagonal extraction: D[m][m] lives at (vgpr m, lane m) for m<8 and
  // (vgpr m-8, lane 16+m) for m>=8 -> active lanes 0..7 and 24..31.
  float d2 = 0.0f;
#pragma unroll
  for (int i = 0; i < 8; ++i)
    if (lane == i || lane == 24 + i) d2 = acc[i];

  bool active = (lane < 8) || (lane >= 24);
  if (active) {
    int m = (lane < 8) ? lane : (lane - 16);
    int s = s0 + m;
    if (s < N) {
      uint32_t pw = packed[s];
      int t = (int)(pw & 0xffffu);
      if ((pw >> 16) == 0) {
        atomicAdd(&dists[t], sqrtf(d2));   // clean: per-class distance sum
      } else {
        atomicAdd(&lossacc[0], d2);        // poison: squared-error sum
        atomicAdd(&lossacc[1], 1.0f);      // poison count
      }
    }
  }
}

__global__ void k_final(const float* __restrict__ centers_new,
                        const float* __restrict__ counts,
                        const float* __restrict__ dists,
                        const float* __restrict__ lossacc,
                        const float* __restrict__ radius,
                        float* __restrict__ out) {
  int i = blockIdx.x * blockDim.x + threadIdx.x;
  if (i < NCLS * DIM) out[1 + i] = centers_new[i];          // center_new
  if (i < NCLS) {
    float cnt = counts[i];
    float r   = radius[i];
    float dm  = dists[i] / fmaxf(cnt, 1.0f);
    out[1 + NCLS * DIM + i] = (cnt > 0.0f) ? (r * MOM + dm * (1.0f - MOM)) : r;
  }
  if (i == 0) {
    float sq = lossacc[0], pc = lossacc[1];
    out[0] = (pc > 0.0f) ? (sq / fmaxf(pc * (float)DIM, 1.0f)) : 0.0f;
  }
}

extern "C" void kernel_launch(void* const* d_in, const int* in_sizes, int n_in,
                              void* d_out, int out_size, void* d_ws, size_t ws_size,
                              hipStream_t stream) {
  const float* features = (const float*)d_in[0];
  const float* center   = (const float*)d_in[1];
  const float* radius   = (const float*)d_in[2];
  const int*   targets  = (const int*)d_in[3];
  const int*   pmarks   = (const int*)d_in[4];
  float* out = (float*)d_out;
  const int N = in_sizes[3];                  // targets count = 262144

  uint32_t* ws_packed  = (uint32_t*)d_ws;
  float*    ws_centers = (float*)d_ws + N;
  float*    ws_counts  = ws_centers + (size_t)NCLS * DIM;
  float*    ws_dist    = ws_counts + NCLS;
  float*    ws_loss    = ws_dist + NCLS;

  k_init<<<(N + 255) / 256, 256, 0, stream>>>(targets, pmarks, ws_packed, ws_dist, ws_loss, N);
  k_center<<<NCLS, 32, 0, stream>>>(ws_packed, features, center, ws_centers, ws_counts, N);

  int tiles  = (N + 15) / 16;
  int blocks = (tiles + K2_WAVES - 1) / K2_WAVES;
  k_dist<<<blocks, 32 * K2_WAVES, 0, stream>>>(ws_packed, features, ws_centers, ws_dist, ws_loss, N);

  k_final<<<(NCLS * DIM + 255) / 256, 256, 0, stream>>>(ws_centers, ws_counts, ws_dist, ws_loss, radius, out);
}